// EceLabelShift_57501022159077
// MI455X (gfx1250) — compile-verified
//
#include <hip/hip_runtime.h>
#include <hip/hip_bf16.h>
#include <stdint.h>
#include <math.h>

#define NCLS 100
#define NBINS 15
#define WPB 8            // waves per block in the streaming kernels
#define TILE_ROWS 16     // one wave owns 16 rows (matches WMMA 16x16x4 A layout)

typedef __attribute__((ext_vector_type(2))) float v2f;
typedef __attribute__((ext_vector_type(8))) float v8f;

struct Ctrl {
  unsigned nslots;
  unsigned sbin[32];      // coarse (high-16) bins that contain needed ranks
  int      rankSlot[32];  // rank j -> slot index into sbin/hist2
  unsigned rankBin[32];   // rank j -> coarse bin
  unsigned rankRes[32];   // rank j -> residual rank inside coarse bin
  float    frac[16];      // interp fraction per interior quantile q=1..14
  float    orderStat[32]; // resolved order statistics
  float    edges[16];     // final equal-mass bin edges
};

struct Accum {
  float    wnum[NBINS];   // einsum('bn,n->b', in_s, w_per)
  float    s1[NBINS];     // sum conf_t per bin
  float    s2[NBINS];     // sum conf_t^2 per bin
  unsigned cnt[NBINS];    // count_t per bin
};

__device__ __forceinline__ unsigned conf_key(float c) {
  // conf > 0 always, so flipping the sign bit makes uint order == float order
  return __float_as_uint(c) ^ 0x80000000u;
}

// ---------------------------------------------------------------------------
// Streaming stats kernel: async-LDS tile staging + wave32 max/argmax +
// WMMA-accumulated softmax denominator.  mode 0 = target (emit conf_t + coarse
// histogram), mode 1 = source (emit weighted-accuracy per equal-mass bin).
// ---------------------------------------------------------------------------
__global__ __launch_bounds__(256) void stats_kernel(
    const float* __restrict__ logits, int nRows, int mode,
    float* __restrict__ conf_out, unsigned* __restrict__ hist1,
    const int* __restrict__ labels, const float* __restrict__ weight,
    const float* __restrict__ edges_g, float* __restrict__ wnum_g)
{
  __shared__ float tile[WPB][TILE_ROWS * NCLS];   // 6400 B per wave, contiguous
  __shared__ float xch[WPB][16];                  // row-sum exchange
  __shared__ float binAcc[NBINS];

  const int lane = threadIdx.x & 31;
  const int wave = threadIdx.x >> 5;

  if (mode == 1 && threadIdx.x < NBINS) binAcc[threadIdx.x] = 0.0f;
  __syncthreads();

  float e[16];
  if (mode == 1) {
#pragma unroll
    for (int j = 0; j < 16; ++j) e[j] = edges_g[j];   // uniform -> scalar loads
  }

  const int nTiles = nRows / TILE_ROWS;
  const int waveId = blockIdx.x * WPB + wave;
  const int nWaves = gridDim.x * WPB;

  const unsigned ldsBase = (unsigned)(unsigned long long)(uintptr_t)(&tile[wave][0]);
  const int r = lane & 15;   // row within tile
  const int h = lane >> 4;   // K-half (WMMA A layout: lanes 16..31 hold K=2,3)

  for (int t = waveId; t < nTiles; t += nWaves) {
    // ---- async stage 6400 contiguous bytes into this wave's LDS tile ----
    unsigned long long g =
        (unsigned long long)(uintptr_t)(logits + (size_t)t * TILE_ROWS * NCLS);
#pragma unroll
    for (int i = 0; i < 12; ++i) {                 // 12 x 32 lanes x 16B = 6144B
      unsigned off = (unsigned)(i * 512 + lane * 16);
      unsigned l = ldsBase + off;
      unsigned long long ga = g + off;
      asm volatile("global_load_async_to_lds_b128 %0, %1, off"
                   :: "v"(l), "v"(ga) : "memory");
    }
    {                                              // remaining 256B: 32 x 8B
      unsigned off = 6144u + (unsigned)lane * 8u;
      unsigned l = ldsBase + off;
      unsigned long long ga = g + off;
      asm volatile("global_load_async_to_lds_b64 %0, %1, off"
                   :: "v"(l), "v"(ga) : "memory");
    }
    asm volatile("s_wait_asynccnt 0" ::: "memory");

    const float* row = &tile[wave][r * NCLS];

    // ---- phase A: max / first-argmax (lane pair splits the 100 classes) ----
    float m = -__builtin_inff();
    int am = 0;
#pragma unroll 10
    for (int k = h * 50; k < h * 50 + 50; ++k) {
      float v = row[k];
      if (v > m) { m = v; am = k; }
    }
    float mo = __shfl_xor(m, 16);
    int   ao = __shfl_xor(am, 16);
    if (mo > m || (mo == m && ao < am)) { m = mo; am = ao; }

    // ---- phase B: sum exp(x - m) via v_wmma_f32_16x16x4_f32, B = ones ----
    // A layout: lane(0..15)=rows, VGPR0/1 = K0/K1; lane(16..31) = K2/K3.
    v8f c = {};
    const v2f ones = {1.0f, 1.0f};
#pragma unroll
    for (int k0 = 0; k0 < NCLS; k0 += 4) {
      float x0 = row[k0 + h * 2 + 0];
      float x1 = row[k0 + h * 2 + 1];
      v2f a;
      a[0] = __expf(x0 - m);     // v_exp_f32 co-executes with the matrix pipe
      a[1] = __expf(x1 - m);
      c = __builtin_amdgcn_wmma_f32_16x16x4_f32(false, a, false, ones,
                                                (short)0, c, false, false);
    }
    // C/D layout: lane0 VGPR j = row j sum, lane16 VGPR j = row j+8 sum
    if (lane == 0) {
#pragma unroll
      for (int j = 0; j < 8; ++j) xch[wave][j] = c[j];
    }
    if (lane == 16) {
#pragma unroll
      for (int j = 0; j < 8; ++j) xch[wave][8 + j] = c[j];
    }

    if (h == 0) {
      float s = xch[wave][r];
      float conf = 1.0f / s;     // max softmax = exp(0)/sumexp
      int rowIdx = t * TILE_ROWS + r;
      if (mode == 0) {
        conf_out[rowIdx] = conf;
        atomicAdd(&hist1[conf_key(conf) >> 16], 1u);
      } else {
        int lab = labels[rowIdx];
        float w = (am == lab) ? weight[lab] : 0.0f;
        if (w != 0.0f) {
          int cgt = 0;
#pragma unroll
          for (int j = 0; j < 16; ++j) cgt += (conf > e[j]) ? 1 : 0;
          if (cgt >= 1 && cgt <= 15) atomicAdd(&binAcc[cgt - 1], w);
        }
      }
    }
  }

  // ---- tail rows (nRows not multiple of 16): plain per-thread path ----
  {
    int rr = nTiles * TILE_ROWS + blockIdx.x * blockDim.x + threadIdx.x;
    if (rr < nRows) {
      const float* rowg = logits + (size_t)rr * NCLS;
      float m = -__builtin_inff(); int am = 0;
      for (int k = 0; k < NCLS; ++k) { float v = rowg[k]; if (v > m) { m = v; am = k; } }
      float s = 0.0f;
      for (int k = 0; k < NCLS; ++k) s += __expf(rowg[k] - m);
      float conf = 1.0f / s;
      if (mode == 0) {
        conf_out[rr] = conf;
        atomicAdd(&hist1[conf_key(conf) >> 16], 1u);
      } else {
        int lab = labels[rr];
        float w = (am == lab) ? weight[lab] : 0.0f;
        if (w != 0.0f) {
          int cgt = 0;
          for (int j = 0; j < 16; ++j) cgt += (conf > e[j]) ? 1 : 0;
          if (cgt >= 1 && cgt <= 15) atomicAdd(&binAcc[cgt - 1], w);
        }
      }
    }
  }

  if (mode == 1) {
    __syncthreads();
    if (threadIdx.x < NBINS) {
      float v = binAcc[threadIdx.x];
      if (v != 0.0f) atomicAdd(&wnum_g[threadIdx.x], v);
    }
  }
}

// ---------------------------------------------------------------------------
// Single block: scan coarse histogram, locate coarse bin + residual for the
// 30 needed order statistics (floor/ceil of each interior quantile + min/max),
// dedupe coarse bins into slots for the refine pass.
// ---------------------------------------------------------------------------
__global__ void plan_kernel(const unsigned* __restrict__ hist1, Ctrl* ctrl, int nT) {
  __shared__ unsigned csum[1025];
  __shared__ unsigned rbin[32];
  int t = threadIdx.x;  // blockDim = 1024, each thread owns 64 coarse bins
  unsigned s = 0;
#pragma unroll 4
  for (int i = 0; i < 64; ++i) s += hist1[t * 64 + i];
  csum[t] = s;
  __syncthreads();
  if (t == 0) {
    unsigned run = 0;
    for (int i = 0; i < 1024; ++i) { unsigned v = csum[i]; csum[i] = run; run += v; }
    csum[1024] = run;
  }
  __syncthreads();

  if (t < 30) {
    unsigned R;
    double x = 0.0;
    if (t == 0) R = 0u;
    else if (t == 29) R = (unsigned)(nT - 1);
    else {
      int q = (t + 1) >> 1;                       // interior quantile 1..14
      x = (double)q * (double)nT / 15.0;          // linspace(0, N, 16)[q]
      unsigned lo = (unsigned)x;                  // floor (positive)
      R = lo + ((t & 1) ? 0u : 1u);               // odd t = floor, even t = ceil
    }
    int lo = 0, hi = 1023;                        // largest chunk with base <= R
    while (lo < hi) {
      int mid = (lo + hi + 1) >> 1;
      if (csum[mid] <= R) lo = mid; else hi = mid - 1;
    }
    unsigned run = csum[lo], bin = 0, res = 0;
    for (int i = 0; i < 64; ++i) {
      unsigned v = hist1[lo * 64 + i];
      if (run + v > R) { bin = (unsigned)(lo * 64 + i); res = R - run; break; }
      run += v;
    }
    rbin[t] = bin;
    ctrl->rankBin[t] = bin;
    ctrl->rankRes[t] = res;
    if (t >= 1 && t <= 28 && (t & 1)) {
      int q = (t + 1) >> 1;
      ctrl->frac[q] = (float)(x - floor(x));
    }
  }
  __syncthreads();
  if (t == 0) {
    int nsl = 0;
    for (int j = 0; j < 30; ++j) {
      unsigned b = rbin[j];
      int slot = -1;
      for (int k = 0; k < nsl; ++k) if (ctrl->sbin[k] == b) { slot = k; break; }
      if (slot < 0) { slot = nsl; ctrl->sbin[nsl++] = b; }
      ctrl->rankSlot[j] = slot;
    }
    ctrl->nslots = (unsigned)nsl;
  }
}

// Refine pass: low-16 histograms restricted to the coarse bins that matter.
__global__ __launch_bounds__(256) void refine_kernel(
    const float* __restrict__ conf_t, int n,
    const Ctrl* __restrict__ ctrl, unsigned* __restrict__ hist2)
{
  __shared__ unsigned sb[32];
  __shared__ unsigned nslS;
  if (threadIdx.x == 0) nslS = ctrl->nslots;
  if (threadIdx.x < 32) sb[threadIdx.x] = ctrl->sbin[threadIdx.x];
  __syncthreads();
  int ns = (int)nslS;
  for (int i = blockIdx.x * blockDim.x + threadIdx.x; i < n;
       i += gridDim.x * blockDim.x) {
    unsigned key = conf_key(conf_t[i]);
    unsigned cb = key >> 16;
    for (int s = 0; s < ns; ++s)
      if (sb[s] == cb) {
        atomicAdd(&hist2[(size_t)s * 65536 + (key & 0xFFFFu)], 1u);
        break;
      }
  }
}

// One warp per rank: scan its slot's low-16 histogram, recover exact key ->
// exact float order statistic; thread 0 then interpolates the 16 edges.
__global__ void edges_kernel(const unsigned* __restrict__ hist2, Ctrl* ctrl) {
  int wid = threadIdx.x >> 5;
  int lane = threadIdx.x & 31;
  if (wid < 30) {
    int slot = ctrl->rankSlot[wid];
    unsigned cb = ctrl->sbin[slot];
    unsigned res = ctrl->rankRes[wid];
    const unsigned* hh = hist2 + (size_t)slot * 65536;
    unsigned ps = 0;
    for (int i = 0; i < 2048; ++i) ps += hh[lane * 2048 + i];
    unsigned incl = ps;
#pragma unroll
    for (int d = 1; d < 32; d <<= 1) {
      unsigned v = __shfl_up(incl, d);
      if (lane >= d) incl += v;
    }
    unsigned ex = incl - ps;
    if (res >= ex && res < ex + ps) {
      unsigned run = ex;
      for (int i = 0; i < 2048; ++i) {
        unsigned v = hh[lane * 2048 + i];
        if (run + v > res) {
          unsigned key = (cb << 16) | (unsigned)(lane * 2048 + i);
          ctrl->orderStat[wid] = __uint_as_float(key ^ 0x80000000u);
          break;
        }
        run += v;
      }
    }
  }
  __syncthreads();
  if (threadIdx.x == 0) {
    ctrl->edges[0]  = ctrl->orderStat[0];
    ctrl->edges[15] = ctrl->orderStat[29];
    for (int q = 1; q <= 14; ++q) {
      float lo = ctrl->orderStat[2 * q - 1];
      float hi = ctrl->orderStat[2 * q];
      ctrl->edges[q] = lo + ctrl->frac[q] * (hi - lo);
    }
  }
}

// Target binning: per-bin moments (count, sum, sumsq) instead of the dense
// [B,Nt] pass: sum (c-ce)^2 = s2 - 2*ce*s1 + n*ce^2.
__global__ __launch_bounds__(256) void tbin_kernel(
    const float* __restrict__ conf_t, int n,
    const Ctrl* __restrict__ ctrl, Accum* acc)
{
  __shared__ float e[16];
  __shared__ float ls1[NBINS], ls2[NBINS];
  __shared__ unsigned lc[NBINS];
  if (threadIdx.x < 16) e[threadIdx.x] = ctrl->edges[threadIdx.x];
  if (threadIdx.x < NBINS) { ls1[threadIdx.x] = 0.f; ls2[threadIdx.x] = 0.f; lc[threadIdx.x] = 0u; }
  __syncthreads();
  for (int i = blockIdx.x * blockDim.x + threadIdx.x; i < n;
       i += gridDim.x * blockDim.x) {
    float c = conf_t[i];
    int cgt = 0;
#pragma unroll
    for (int j = 0; j < 16; ++j) cgt += (c > e[j]) ? 1 : 0;
    if (cgt >= 1 && cgt <= 15) {
      int b = cgt - 1;
      atomicAdd(&lc[b], 1u);
      atomicAdd(&ls1[b], c);
      atomicAdd(&ls2[b], c * c);
    }
  }
  __syncthreads();
  if (threadIdx.x < NBINS) {
    if (lc[threadIdx.x])          atomicAdd(&acc->cnt[threadIdx.x], lc[threadIdx.x]);
    if (ls1[threadIdx.x] != 0.f)  atomicAdd(&acc->s1[threadIdx.x], ls1[threadIdx.x]);
    if (ls2[threadIdx.x] != 0.f)  atomicAdd(&acc->s2[threadIdx.x], ls2[threadIdx.x]);
  }
}

__global__ void final_kernel(const Accum* __restrict__ acc, float* out,
                             int nS, int nT) {
  if (threadIdx.x == 0 && blockIdx.x == 0) {
    float normalizer = (float)(nT - 1) / (float)nS;
    float total = 0.f;
    for (int b = 0; b < NBINS; ++b) {
      unsigned c = acc->cnt[b];
      float denom = (float)((c > 1u) ? (c - 1u) : 1u);
      float ce = normalizer * acc->wnum[b] / denom;
      float pb = acc->s2[b] - 2.f * ce * acc->s1[b] + (float)c * ce * ce;
      if (c > 1u) total += pb;
    }
    out[0] = total / (float)nT;
  }
}

extern "C" void kernel_launch(void* const* d_in, const int* in_sizes, int n_in,
                              void* d_out, int out_size, void* d_ws, size_t ws_size,
                              hipStream_t stream) {
  (void)n_in; (void)out_size; (void)ws_size;
  const float* logits_s = (const float*)d_in[0];
  const int*   labels_s = (const int*)d_in[1];
  const float* logits_t = (const float*)d_in[2];
  const float* weight   = (const float*)d_in[3];
  const int nS = in_sizes[0] / NCLS;
  const int nT = in_sizes[2] / NCLS;

  char* ws = (char*)d_ws;
  size_t off = 0;
  float* conf_t = (float*)(ws + off);
  off += ((size_t)nT * 4 + 255) & ~(size_t)255;
  size_t zero_from = off;
  unsigned* hist1 = (unsigned*)(ws + off); off += (size_t)65536 * 4;
  unsigned* hist2 = (unsigned*)(ws + off); off += (size_t)32 * 65536 * 4;
  Ctrl*  ctrl = (Ctrl*)(ws + off);  off += (sizeof(Ctrl)  + 255) & ~(size_t)255;
  Accum* acc  = (Accum*)(ws + off); off += (sizeof(Accum) + 255) & ~(size_t)255;

  hipMemsetAsync(ws + zero_from, 0, off - zero_from, stream);

  dim3 blk(256);
  // 1) target: conf_t + coarse histogram (streams 200 MB, HBM-bound)
  stats_kernel<<<dim3(1024), blk, 0, stream>>>(logits_t, nT, 0, conf_t, hist1,
                                               nullptr, nullptr, nullptr, nullptr);
  // 2) locate the 30 needed order statistics in coarse bins
  plan_kernel<<<dim3(1), dim3(1024), 0, stream>>>(hist1, ctrl, nT);
  // 3) low-16 histograms for the hit bins (reads 2 MB)
  refine_kernel<<<dim3(1024), blk, 0, stream>>>(conf_t, nT, ctrl, hist2);
  // 4) exact order statistics -> equal-mass edges
  edges_kernel<<<dim3(1), dim3(1024), 0, stream>>>(hist2, ctrl);
  // 5) source: conf/argmax/weighted-acc fused with binning (streams 200 MB)
  stats_kernel<<<dim3(1024), blk, 0, stream>>>(logits_s, nS, 1, nullptr, nullptr,
                                               labels_s, weight, ctrl->edges,
                                               &acc->wnum[0]);
  // 6) target per-bin moments (reads 2 MB)
  tbin_kernel<<<dim3(1024), blk, 0, stream>>>(conf_t, nT, ctrl, acc);
  // 7) reduce 15 bins -> scalar
  final_kernel<<<dim3(1), dim3(32), 0, stream>>>(acc, (float*)d_out, nS, nT);
}